// Graph_DNN_1589137899718
// MI455X (gfx1250) — compile-verified
//
#include <hip/hip_runtime.h>

// Problem constants (match the reference harness; scalars pred_len/input_len
// are fixed by the reference file, so hardcode for host-side grid sizing).
#define N_NODES   (32 * 4096)   // 131072
#define NUM_EDGES 4194304
#define F_IN      256
#define HID       3
#define INPUT_LEN 4096
#define PRED_LEN  1024

typedef __attribute__((ext_vector_type(2))) float v2f;
typedef __attribute__((ext_vector_type(8))) float v8f;

// ---------------------------------------------------------------------------
// ws layout (floats):
//   dinv  : [0,        N)        degree -> deg^-1/2 in place
//   hlin  : [N,        5N)       x @ W1, stride 4 per node (cols 0..2 used)
//   acc1  : [5N,       9N)       layer-1 edge accumulator, stride 4
//   acc2  : [9N,      10N)       layer-2 edge accumulator
//   h2lin : [10N,     11N)       relu(h1) @ W2 per node
// ---------------------------------------------------------------------------

__global__ void init_deg_kernel(float* __restrict__ deg) {
    int i = blockIdx.x * 256 + threadIdx.x;
    if (i < N_NODES) deg[i] = 1.0f;            // self-loop folded in
}

__global__ void zero_kernel(float* __restrict__ p, int n) {
    int i = blockIdx.x * 256 + threadIdx.x;
    if (i < n) p[i] = 0.0f;
}

__global__ void degree_kernel(const int* __restrict__ dst,
                              float* __restrict__ deg) {
    int e = blockIdx.x * 256 + threadIdx.x;
    if (e < NUM_EDGES) atomicAdd(&deg[dst[e]], 1.0f);
}

__global__ void dinv_kernel(float* __restrict__ deg) {
    int i = blockIdx.x * 256 + threadIdx.x;
    if (i < N_NODES) deg[i] = __frsqrt_rn(deg[i]);   // deg >= 1 always
}

// ---------------------------------------------------------------------------
// Layer-1 linear: hlin[N,3] = x[N,256] @ W1[256,3], via V_WMMA_F32_16X16X4_F32.
// One wave computes one 16-row tile, iterating K in steps of 4.
// W1 is zero-padded to 256x16 in LDS.
// A frag  (16x4 f32, 2 VGPR): lanes 0-15 -> M=lane, K=k..k+1
//                             lanes 16-31 -> M=lane-16, K=k+2..k+3
// B frag  (4x16 f32, 2 VGPR): same K split across wave halves, N=lane&15
// C/D     (16x16 f32, 8 VGPR): c[r] = (M = r + 8*(lane>>4), N = lane&15)
// ---------------------------------------------------------------------------
__global__ void __launch_bounds__(256)
gemm1_wmma_kernel(const float* __restrict__ x,
                  const float* __restrict__ W1,
                  float* __restrict__ hlin) {
    __shared__ float Wpad[F_IN][16];
    for (int i = threadIdx.x; i < F_IN * 16; i += 256) {
        int k = i >> 4, n = i & 15;
        Wpad[k][n] = (n < HID) ? W1[k * HID + n] : 0.0f;
    }
    __syncthreads();

    const int wave = threadIdx.x >> 5;
    const int lane = threadIdx.x & 31;
    const int half = lane >> 4;          // 0: K=k..k+1, 1: K=k+2..k+3
    const int col  = lane & 15;          // A row within tile / B,N column
    const int tileBase = (blockIdx.x * 8 + wave) * 16;

    const float* xrow = x + (size_t)(tileBase + col) * F_IN;

    v8f c = {};
    #pragma unroll 4
    for (int k = 0; k < F_IN; k += 4) {
        const int kk = k + 2 * half;
        v2f a = *(const v2f*)(xrow + kk);           // 8B, 8B-aligned
        v2f b;
        b.x = Wpad[kk][col];
        b.y = Wpad[kk + 1][col];
        c = __builtin_amdgcn_wmma_f32_16x16x4_f32(
                /*neg_a=*/false, a, /*neg_b=*/false, b,
                /*c_mod=*/(short)0, c, /*reuse_a=*/false, /*reuse_b=*/false);
    }

    if (col < HID) {
        #pragma unroll
        for (int r = 0; r < 8; ++r) {
            int row = tileBase + r + 8 * half;
            hlin[(size_t)row * 4 + col] = c[r];
        }
    }
}

// Layer-1 edge scatter: acc1[dst] += hlin[src] * dinv[src] * dinv[dst]
__global__ void aggregate1_kernel(const int* __restrict__ src,
                                  const int* __restrict__ dst,
                                  const float* __restrict__ dinv,
                                  const float* __restrict__ hlin,
                                  float* __restrict__ acc1) {
    int e = blockIdx.x * 256 + threadIdx.x;
    if (e >= NUM_EDGES) return;
    int s = src[e], d = dst[e];
    float nrm = dinv[s] * dinv[d];
    const float4 h = *(const float4*)(hlin + (size_t)s * 4);   // 16B aligned
    atomicAdd(&acc1[(size_t)d * 4 + 0], h.x * nrm);
    atomicAdd(&acc1[(size_t)d * 4 + 1], h.y * nrm);
    atomicAdd(&acc1[(size_t)d * 4 + 2], h.z * nrm);
}

// Fused: add self-loop + bias, ReLU, then the tiny 3->1 GEMM (layer-2 linear).
__global__ void finalize1_kernel(const float* __restrict__ dinv,
                                 const float* __restrict__ hlin,
                                 const float* __restrict__ acc1,
                                 const float* __restrict__ b1,
                                 const float* __restrict__ W2,
                                 float* __restrict__ h2lin) {
    int i = blockIdx.x * 256 + threadIdx.x;
    if (i >= N_NODES) return;
    float di2 = dinv[i] * dinv[i];
    float out2 = 0.0f;
    #pragma unroll
    for (int cidx = 0; cidx < HID; ++cidx) {
        float v = acc1[(size_t)i * 4 + cidx]
                + hlin[(size_t)i * 4 + cidx] * di2 + b1[cidx];
        v = fmaxf(v, 0.0f);                 // ReLU
        out2 = fmaf(v, W2[cidx], out2);
    }
    h2lin[i] = out2;
}

// Layer-2 edge scatter, filtered: only destinations that land in the
// [:, :PRED_LEN] output window matter (drops 75% of the atomics).
__global__ void aggregate2_kernel(const int* __restrict__ src,
                                  const int* __restrict__ dst,
                                  const float* __restrict__ dinv,
                                  const float* __restrict__ h2lin,
                                  float* __restrict__ acc2) {
    int e = blockIdx.x * 256 + threadIdx.x;
    if (e >= NUM_EDGES) return;
    int d = dst[e];
    if ((d & (INPUT_LEN - 1)) >= PRED_LEN) return;
    int s = src[e];
    atomicAdd(&acc2[d], h2lin[s] * dinv[s] * dinv[d]);
}

// Gather the [B, PRED_LEN, 1] output window, add self-loop + bias b2.
__global__ void output_kernel(const float* __restrict__ dinv,
                              const float* __restrict__ h2lin,
                              const float* __restrict__ acc2,
                              const float* __restrict__ b2,
                              float* __restrict__ out) {
    int i = blockIdx.x * 256 + threadIdx.x;      // i in [0, B*PRED_LEN)
    if (i >= 32 * PRED_LEN) return;
    int b = i / PRED_LEN;
    int t = i - b * PRED_LEN;
    int node = b * INPUT_LEN + t;
    float di2 = dinv[node] * dinv[node];
    out[i] = acc2[node] + h2lin[node] * di2 + b2[0];
}

extern "C" void kernel_launch(void* const* d_in, const int* in_sizes, int n_in,
                              void* d_out, int out_size, void* d_ws, size_t ws_size,
                              hipStream_t stream) {
    const float* x  = (const float*)d_in[0];   // [N, 256]
    const float* W1 = (const float*)d_in[1];   // [256, 3]
    const float* b1 = (const float*)d_in[2];   // [3]
    const float* W2 = (const float*)d_in[3];   // [3, 1]
    const float* b2 = (const float*)d_in[4];   // [1]
    const int*   ei = (const int*)d_in[5];     // [2, E] flat
    const int* src = ei;
    const int* dst = ei + NUM_EDGES;

    float* ws    = (float*)d_ws;
    float* dinv  = ws;                         // N
    float* hlin  = ws + (size_t)1 * N_NODES;   // 4N
    float* acc1  = ws + (size_t)5 * N_NODES;   // 4N
    float* acc2  = ws + (size_t)9 * N_NODES;   // N   (contiguous with acc1)
    float* h2lin = ws + (size_t)10 * N_NODES;  // N

    float* outp = (float*)d_out;

    const int T = 256;
    // 1) init: deg=1 (self-loops), zero acc1+acc2 (5N contiguous floats)
    init_deg_kernel<<<N_NODES / T, T, 0, stream>>>(dinv);
    zero_kernel<<<(5 * N_NODES) / T, T, 0, stream>>>(acc1, 5 * N_NODES);
    // 2) degree over destinations
    degree_kernel<<<NUM_EDGES / T, T, 0, stream>>>(dst, dinv);
    // 3) deg -> deg^{-1/2}
    dinv_kernel<<<N_NODES / T, T, 0, stream>>>(dinv);
    // 4) x @ W1 via f32 WMMA (one 16-row tile per wave, 8 waves/block)
    gemm1_wmma_kernel<<<N_NODES / (16 * 8), T, 0, stream>>>(x, W1, hlin);
    // 5) layer-1 edge scatter
    aggregate1_kernel<<<NUM_EDGES / T, T, 0, stream>>>(src, dst, dinv, hlin, acc1);
    // 6) self-loop + bias + ReLU + (3->1) linear
    finalize1_kernel<<<N_NODES / T, T, 0, stream>>>(dinv, hlin, acc1, b1, W2, h2lin);
    // 7) layer-2 edge scatter (filtered to output window)
    aggregate2_kernel<<<NUM_EDGES / T, T, 0, stream>>>(src, dst, dinv, h2lin, acc2);
    // 8) gather output window
    output_kernel<<<(32 * PRED_LEN) / T, T, 0, stream>>>(dinv, h2lin, acc2, b2, outp);
}